// PointerGenerator_63161789055659
// MI455X (gfx1250) — compile-verified
//
#include <hip/hip_runtime.h>
#include <hip/hip_bf16.h>

// ---------------------------------------------------------------------------
// PointerGenerator fused implementation for gfx1250 (MI455X).
//
// Sizes: B=16, TQ=64, TK=512, D=256, V=32000.
// Output (B,TQ,V) f32 = 131 MB -> memory-bound (~5.6us floor @ 23.3 TB/s).
// Vocab GEMM: 2*1024*32000*256 = 16.8 GFLOP f32 -> trivial on WMMA pipe.
// W_state = 32.8 MB -> resident in 192 MB L2, so re-streaming per pass is cheap.
// All GEMMs use V_WMMA_F32_16X16X4_F32 (keeps f32 precision for softmax+log).
// ---------------------------------------------------------------------------

#define B_  16
#define TQ_ 64
#define TK_ 512
#define D_  256
#define V_  32000

#define VB1 2048            // vocab block for the stats pass
#define NVB1 16             // ceil(32000/2048): 15 full + 1280
#define VB2 512             // vocab block for the final pass (fits 16x512 f32 LDS tile)
#define NVB2 63             // ceil(32000/512): 62 full + 256

typedef __attribute__((ext_vector_type(2))) float v2f;
typedef __attribute__((ext_vector_type(8))) float v8f;

// D(16x16,f32) = A(16x4,f32) x B(4x16,f32) + C
// A layout: lanes 0-15 rows M, vgpr{0,1} = K{0,1}; lanes 16-31 = K{2,3}
// C layout: vgpr r: lane<16 -> (M=r, N=lane); lane>=16 -> (M=8+r, N=lane-16)
__device__ inline v8f wmma_f32_k4(v2f a, v2f b, v8f c) {
  return __builtin_amdgcn_wmma_f32_16x16x4_f32(
      /*neg_a=*/false, a, /*neg_b=*/false, b,
      /*c_mod=*/(short)0, c, /*reuse_a=*/false, /*reuse_b=*/false);
}

__device__ inline float wave_max(float x) {
  #pragma unroll
  for (int o = 1; o < 32; o <<= 1) x = fmaxf(x, __shfl_xor(x, o, 32));
  return x;
}
__device__ inline float wave_sum(float x) {
  #pragma unroll
  for (int o = 1; o < 32; o <<= 1) x += __shfl_xor(x, o, 32);
  return x;
}
// reductions across the 16 lanes of a half-wave (WMMA C layout halves)
__device__ inline float half16_max(float x) {
  #pragma unroll
  for (int o = 1; o < 16; o <<= 1) x = fmaxf(x, __shfl_xor(x, o, 32));
  return x;
}
__device__ inline float half16_sum(float x) {
  #pragma unroll
  for (int o = 1; o < 16; o <<= 1) x += __shfl_xor(x, o, 32);
  return x;
}

// ---------------------------------------------------------------------------
// K1: per (b, 16-query tile): attention scores (WMMA), softmax, context_vec
//     (WMMA), p_gen sigmoid. 64 workgroups x 256 threads (8 waves).
// ---------------------------------------------------------------------------
__global__ __launch_bounds__(256) void k_attn(
    const float* __restrict__ dom, const float* __restrict__ out_st,
    const float* __restrict__ ctx, const float* __restrict__ Wg,
    const float* __restrict__ bg, float* __restrict__ attnW,
    float* __restrict__ pgenW)
{
  __shared__ float sS[16][TK_];   // 32 KB: scores -> attention probs
  __shared__ float sCV[16][D_];   // 16 KB: context vector

  const int b    = blockIdx.x >> 2;
  const int q0   = (blockIdx.x & 3) << 4;
  const int wave = threadIdx.x >> 5;
  const int lane = threadIdx.x & 31;
  const int row  = lane & 15;
  const int hi   = lane >> 4;

  // ---- scores S[q,k] = sum_d out[q,d]*ctx[k,d], 4 k-tiles per wave
  v8f acc[4];
  #pragma unroll
  for (int i = 0; i < 4; i++) acc[i] = {};
  const float* Abase = out_st + (size_t)(b*TQ_ + q0 + row)*D_;
  for (int dc = 0; dc < D_/4; ++dc) {
    const int d0 = dc*4;
    v2f a = *(const v2f*)(Abase + d0 + 2*hi);
    #pragma unroll
    for (int i = 0; i < 4; i++) {
      const int k0 = (wave*4 + i)*16;
      // B[kk][n] = ctx[k0+n][d0+kk]; lane half selects kk pair
      v2f bb = *(const v2f*)(ctx + (size_t)(b*TK_ + k0 + row)*D_ + d0 + 2*hi);
      acc[i] = wmma_f32_k4(a, bb, acc[i]);
    }
  }
  #pragma unroll
  for (int i = 0; i < 4; i++) {
    const int k0 = (wave*4 + i)*16;
    #pragma unroll
    for (int r = 0; r < 8; r++)
      sS[hi*8 + r][k0 + row] = acc[i][r] * 0.0625f;   // 1/sqrt(256)
  }
  __syncthreads();

  // ---- softmax over TK per row (2 rows per wave)
  #pragma unroll
  for (int j = 0; j < 2; j++) {
    const int q = wave*2 + j;
    float v[TK_/32];
    float m = -1e30f;
    #pragma unroll
    for (int t = 0; t < TK_/32; t++) { v[t] = sS[q][lane + 32*t]; m = fmaxf(m, v[t]); }
    m = wave_max(m);
    float s = 0.f;
    #pragma unroll
    for (int t = 0; t < TK_/32; t++) { v[t] = __expf(v[t] - m); s += v[t]; }
    s = wave_sum(s);
    const float inv = 1.0f / s;
    #pragma unroll
    for (int t = 0; t < TK_/32; t++) {
      const float p = v[t] * inv;
      sS[q][lane + 32*t] = p;
      attnW[(size_t)(b*TQ_ + q0 + q)*TK_ + lane + 32*t] = p;
    }
  }
  __syncthreads();

  // ---- context_vec[q,d] = sum_k attn[q,k]*ctx[k,d] (2 d-tiles per wave)
  v8f cv[2]; cv[0] = {}; cv[1] = {};
  for (int kc = 0; kc < TK_/4; ++kc) {
    const int k0 = kc*4;
    v2f a;
    a.x = sS[row][k0 + 2*hi];
    a.y = sS[row][k0 + 2*hi + 1];
    #pragma unroll
    for (int i = 0; i < 2; i++) {
      const int n0 = (wave*2 + i)*16;
      v2f bb;
      bb.x = ctx[(size_t)(b*TK_ + k0 + 2*hi    )*D_ + n0 + row];
      bb.y = ctx[(size_t)(b*TK_ + k0 + 2*hi + 1)*D_ + n0 + row];
      cv[i] = wmma_f32_k4(a, bb, cv[i]);
    }
  }
  #pragma unroll
  for (int i = 0; i < 2; i++) {
    const int n0 = (wave*2 + i)*16;
    #pragma unroll
    for (int r = 0; r < 8; r++)
      sCV[hi*8 + r][n0 + row] = cv[i][r];
  }
  __syncthreads();

  // ---- p_gen = sigmoid([out_st, ctx_vec, dom] . W_gen + b_gen)
  #pragma unroll
  for (int j = 0; j < 2; j++) {
    const int q = wave*2 + j;
    float dot = 0.f;
    for (int d = lane; d < D_; d += 32) {
      dot += out_st[(size_t)(b*TQ_ + q0 + q)*D_ + d] * Wg[d];
      dot += sCV[q][d]                               * Wg[D_ + d];
      dot += dom[(size_t)(b*TQ_ + q0 + q)*D_ + d]    * Wg[2*D_ + d];
    }
    dot = wave_sum(dot);
    if (lane == 0) {
      const float x = dot + bg[0];
      pgenW[b*TQ_ + q0 + q] = 1.0f / (1.0f + __expf(-x));
    }
  }
}

// ---------------------------------------------------------------------------
// K2: vocab-logit stats. Per (b, q-tile, 2048-col vocab block): compute logit
//     tiles via WMMA, online softmax (running max/sum) per row, write partials.
// ---------------------------------------------------------------------------
__global__ __launch_bounds__(256) void k_stats(
    const float* __restrict__ out_st, const float* __restrict__ Ws,
    const float* __restrict__ bs, float* __restrict__ pmW,
    float* __restrict__ psW)
{
  const int b    = blockIdx.x >> 2;
  const int q0   = (blockIdx.x & 3) << 4;
  const int vb   = blockIdx.y;
  const int v0   = vb * VB1;
  const int cols = min(VB1, V_ - v0);
  const int ntiles = cols >> 4;          // 128 or 80 (V % 16 == 0)

  const int wave = threadIdx.x >> 5;
  const int lane = threadIdx.x & 31;
  const int row  = lane & 15;
  const int hi   = lane >> 4;

  __shared__ float sM[8][16];
  __shared__ float sSm[8][16];

  float rm[8], rs[8];
  #pragma unroll
  for (int r = 0; r < 8; r++) { rm[r] = -1e30f; rs[r] = 0.f; }

  const float* Abase = out_st + (size_t)(b*TQ_ + q0 + row)*D_;
  for (int nt = wave; nt < ntiles; nt += 8) {
    const int c0 = v0 + nt*16;
    v8f acc = {};
    for (int dc = 0; dc < D_/4; ++dc) {
      const int d0 = dc*4;
      v2f a = *(const v2f*)(Abase + d0 + 2*hi);
      v2f bb;                             // B[kk][n] = W_state[d0+kk][c0+n]
      bb.x = Ws[(size_t)(d0 + 2*hi    )*V_ + c0 + row];
      bb.y = Ws[(size_t)(d0 + 2*hi + 1)*V_ + c0 + row];
      acc = wmma_f32_k4(a, bb, acc);
    }
    const float bsv = bs[c0 + row];
    #pragma unroll
    for (int r = 0; r < 8; r++) {
      const float z  = acc[r] + bsv;
      const float tm = half16_max(z);
      const float nm = fmaxf(rm[r], tm);
      const float ts = half16_sum(__expf(z - nm));
      rs[r] = rs[r] * __expf(rm[r] - nm) + ts;
      rm[r] = nm;
    }
  }
  if (row == 0) {           // lane 0 -> rows 0..7, lane 16 -> rows 8..15
    #pragma unroll
    for (int r = 0; r < 8; r++) { sM[wave][hi*8 + r] = rm[r]; sSm[wave][hi*8 + r] = rs[r]; }
  }
  __syncthreads();
  if (threadIdx.x < 16) {
    const int q = threadIdx.x;
    float gm = -1e30f;
    #pragma unroll
    for (int w = 0; w < 8; w++) gm = fmaxf(gm, sM[w][q]);
    float gs = 0.f;
    #pragma unroll
    for (int w = 0; w < 8; w++) gs += sSm[w][q] * __expf(sM[w][q] - gm);
    const int rowid = b*TQ_ + q0 + q;
    pmW[(size_t)rowid*NVB1 + vb] = gm;
    psW[(size_t)rowid*NVB1 + vb] = gs;
  }
}

// ---------------------------------------------------------------------------
// K2b: combine per-block (max,sum) partials -> global row max / sum.
// ---------------------------------------------------------------------------
__global__ __launch_bounds__(256) void k_reduce(
    const float* __restrict__ pmW, const float* __restrict__ psW,
    float* __restrict__ rmW, float* __restrict__ rsW)
{
  const int rowid = blockIdx.x*256 + threadIdx.x;
  if (rowid >= B_*TQ_) return;
  float gm = -1e30f;
  #pragma unroll
  for (int v = 0; v < NVB1; ++v) gm = fmaxf(gm, pmW[(size_t)rowid*NVB1 + v]);
  float gs = 0.f;
  #pragma unroll
  for (int v = 0; v < NVB1; ++v)
    gs += psW[(size_t)rowid*NVB1 + v] * __expf(pmW[(size_t)rowid*NVB1 + v] - gm);
  rmW[rowid] = gm;
  rsW[rowid] = gs;
}

// ---------------------------------------------------------------------------
// K3: final single write pass over the 131 MB output. Recompute logit tiles
//     (W_state served from L2), dense LDS scatter of pointer-attention for
//     this 512-col vocab block, then out = log(pgen*exp(z-m)/s + ptr).
// ---------------------------------------------------------------------------
__global__ __launch_bounds__(256) void k_final(
    const float* __restrict__ out_st, const float* __restrict__ Ws,
    const float* __restrict__ bs, const int* __restrict__ cp,
    const float* __restrict__ attnW, const float* __restrict__ pgenW,
    const float* __restrict__ rmW, const float* __restrict__ rsW,
    float* __restrict__ outp)
{
  const int b    = blockIdx.x >> 2;
  const int q0   = (blockIdx.x & 3) << 4;
  const int vb   = blockIdx.y;
  const int v0   = vb * VB2;
  const int cols = min(VB2, V_ - v0);
  const int ntiles = cols >> 4;          // 32 or 16

  const int wave = threadIdx.x >> 5;
  const int lane = threadIdx.x & 31;
  const int row  = lane & 15;
  const int hi   = lane >> 4;

  __shared__ float sPtr[16][VB2];        // 32 KB dense p_ptr tile
  __shared__ float sScale[16], sMr[16], sOmp[16];

  for (int i = threadIdx.x; i < 16*VB2; i += 256)
    (&sPtr[0][0])[i] = 0.f;
  if (threadIdx.x < 16) {
    const int rowid = b*TQ_ + q0 + threadIdx.x;
    const float pg = pgenW[rowid];
    sScale[threadIdx.x] = pg / rsW[rowid];
    sMr[threadIdx.x]    = rmW[rowid];
    sOmp[threadIdx.x]   = 1.0f - pg;
  }
  __syncthreads();

  // sparse scatter: keys falling in [v0, v0+cols) add attention mass
  for (int k = threadIdx.x; k < TK_; k += 256) {
    const int key = cp[b*TK_ + k];
    if (key >= v0 && key < v0 + cols) {
      const int c = key - v0;
      #pragma unroll
      for (int q = 0; q < 16; q++) {
        const float w = sOmp[q] * attnW[(size_t)(b*TQ_ + q0 + q)*TK_ + k];
        atomicAdd(&sPtr[q][c], w);       // ds_add_f32
      }
    }
  }
  __syncthreads();

  const float* Abase = out_st + (size_t)(b*TQ_ + q0 + row)*D_;
  for (int nt = wave; nt < ntiles; nt += 8) {
    const int c0 = v0 + nt*16;
    v8f acc = {};
    for (int dc = 0; dc < D_/4; ++dc) {
      const int d0 = dc*4;
      v2f a = *(const v2f*)(Abase + d0 + 2*hi);
      v2f bb;
      bb.x = Ws[(size_t)(d0 + 2*hi    )*V_ + c0 + row];
      bb.y = Ws[(size_t)(d0 + 2*hi + 1)*V_ + c0 + row];
      acc = wmma_f32_k4(a, bb, acc);
    }
    const float bsv = bs[c0 + row];
    #pragma unroll
    for (int r = 0; r < 8; r++) {
      const int q = hi*8 + r;
      const float z = acc[r] + bsv;
      const float p = sScale[q] * __expf(z - sMr[q]) + sPtr[q][nt*16 + row];
      outp[(size_t)(b*TQ_ + q0 + q)*V_ + c0 + row] = __logf(p);
    }
  }
}

// ---------------------------------------------------------------------------
extern "C" void kernel_launch(void* const* d_in, const int* in_sizes, int n_in,
                              void* d_out, int out_size, void* d_ws, size_t ws_size,
                              hipStream_t stream) {
  (void)in_sizes; (void)n_in; (void)out_size; (void)ws_size;
  const float* dom    = (const float*)d_in[0];   // (B,TQ,D)
  const float* out_st = (const float*)d_in[1];   // (B,TQ,D)
  const float* ctx    = (const float*)d_in[2];   // (B,TK,D)
  const int*   cp     = (const int*)d_in[3];     // (B,TK) int32
  const float* Ws     = (const float*)d_in[4];   // (D,V)
  const float* bsv    = (const float*)d_in[5];   // (V)
  const float* Wg     = (const float*)d_in[6];   // (3D,1)
  const float* bg     = (const float*)d_in[7];   // (1)
  float* outp = (float*)d_out;                   // (B,TQ,V)

  float* attnW = (float*)d_ws;                           // B*TQ*TK
  float* pgenW = attnW + (size_t)B_*TQ_*TK_;             // B*TQ
  float* pmW   = pgenW + B_*TQ_;                         // B*TQ*NVB1
  float* psW   = pmW + (size_t)B_*TQ_*NVB1;              // B*TQ*NVB1
  float* rmW   = psW + (size_t)B_*TQ_*NVB1;              // B*TQ
  float* rsW   = rmW + B_*TQ_;                           // B*TQ

  k_attn  <<<dim3(B_*TQ_/16),        256, 0, stream>>>(dom, out_st, ctx, Wg, bg, attnW, pgenW);
  k_stats <<<dim3(B_*TQ_/16, NVB1),  256, 0, stream>>>(out_st, Ws, bsv, pmW, psW);
  k_reduce<<<dim3((B_*TQ_+255)/256), 256, 0, stream>>>(pmW, psW, rmW, rsW);
  k_final <<<dim3(B_*TQ_/16, NVB2),  256, 0, stream>>>(out_st, Ws, bsv, cp, attnW, pgenW, rmW, rsW, outp);
}